// ExternalKnowledge_47150150975594
// MI455X (gfx1250) — compile-verified
//
#include <hip/hip_runtime.h>

// Problem constants (from reference)
#define BB   16
#define MM   2048
#define SS   6
#define EE   128
#define LCC  512
#define VV   32000

typedef __attribute__((ext_vector_type(2))) float v2f;
typedef __attribute__((ext_vector_type(8))) float v8f;

// ---------------------------------------------------------------------------
// Kernel 1: base[b,m,:] = sum_s C_k[story[b,m,s]] (+ dh add_lm)
// one wave (32 lanes) per (b,m) row; lane handles a float4 (128 floats/row)
// ---------------------------------------------------------------------------
__global__ void emb_addlm_kernel(const int* __restrict__ story,
                                 const int* __restrict__ kb_len,
                                 const int* __restrict__ conv_len,
                                 const float* __restrict__ dh,
                                 const float* __restrict__ Ck,
                                 float* __restrict__ base) {
  int w    = threadIdx.x >> 5;
  int lane = threadIdx.x & 31;
  int rid  = blockIdx.x * 8 + w;          // [0, B*M)
  int b    = rid >> 11;                   // M == 2048
  int m    = rid & (MM - 1);
  float4 acc = make_float4(0.f, 0.f, 0.f, 0.f);
  const int* st = story + (size_t)rid * SS;
#pragma unroll
  for (int s = 0; s < SS; ++s) {
    int idx = st[s];
    float4 v = ((const float4*)(Ck + (size_t)idx * EE))[lane];
    acc.x += v.x; acc.y += v.y; acc.z += v.z; acc.w += v.w;
  }
  int j = m - kb_len[b];
  if (j >= 0 && j < conv_len[b]) {
    float4 v = ((const float4*)(dh + ((size_t)b * LCC + j) * EE))[lane];
    acc.x += v.x; acc.y += v.y; acc.z += v.z; acc.w += v.w;
  }
  ((float4*)(base + (size_t)rid * EE))[lane] = acc;
}

// ---------------------------------------------------------------------------
// Kernel 2: fold tf half of concat into a per-batch bias:
//   out[b,n] = b1[n] + sum_k tf[b,k] * W1[128+k, n]
// ---------------------------------------------------------------------------
__global__ void tf_fold_kernel(const float* __restrict__ tf,
                               const float* __restrict__ w1,
                               const float* __restrict__ b1,
                               float* __restrict__ out) {
  int t = blockIdx.x * blockDim.x + threadIdx.x;   // [0, B*E)
  int b = t >> 7, n = t & 127;
  float acc = b1[n];
  const float* trow = tf + (size_t)b * EE;
  for (int k = 0; k < EE; ++k) acc += trow[k] * w1[(size_t)(EE + k) * EE + n];
  out[t] = acc;
}

// ---------------------------------------------------------------------------
// Kernel 3: WMMA MLP  Y = lrelu(X @ W1top + tffold[b]) @ W2 + b2
//   X: [B*M, 128], W1top/W2: [128,128] row-major, Y: [B*M, 128]
// 128 threads = 4 waves; block handles 64 rows; wave owns a 16-row strip.
// f32 WMMA 16x16x4, K=128 -> 32 chained wmma per 16x16 tile.
// ---------------------------------------------------------------------------
#define MLP_WAVES 4
#define SW_STRIDE 136   // padded row stride (floats) for weight tile in LDS
#define SX_STRIDE 132   // padded row stride (floats) for X / H tiles in LDS

__global__ void mlp_wmma_kernel(const float* __restrict__ X,
                                const float* __restrict__ w1,
                                const float* __restrict__ tffold,
                                const float* __restrict__ w2,
                                const float* __restrict__ b2,
                                float* __restrict__ Y) {
  extern __shared__ float smem[];
  float* sW = smem;                                  // 128 x SW_STRIDE
  float* sX = sW + EE * SW_STRIDE;                   // MLP_WAVES x 16 x SX_STRIDE
  float* sH = sX + MLP_WAVES * 16 * SX_STRIDE;       // MLP_WAVES x 16 x SX_STRIDE

  const int tid  = threadIdx.x;
  const int w    = tid >> 5;
  const int lane = tid & 31;
  const int half = lane >> 4;       // 0: lanes 0-15, 1: lanes 16-31
  const int m    = lane & 15;

  const size_t blockRow = (size_t)blockIdx.x * (MLP_WAVES * 16);
  const int b  = (int)(blockRow >> 11);              // rows never straddle b
  const size_t r0 = blockRow + (size_t)w * 16;

  // stage W1top (rows 0..127 of [256,128] W1) into LDS
  for (int i = tid; i < EE * EE; i += blockDim.x) {
    int r = i >> 7, c = i & 127;
    sW[r * SW_STRIDE + c] = w1[i];
  }
  // stage this wave's 16 X rows into LDS (coalesced float4)
  float* sXw = sX + w * 16 * SX_STRIDE;
  {
    const float4* Xg = (const float4*)(X + r0 * EE);
    for (int i = lane; i < 16 * (EE / 4); i += 32) {
      int r = i >> 5, c4 = i & 31;
      ((float4*)(sXw + r * SX_STRIDE))[c4] = Xg[r * (EE / 4) + c4];
    }
  }
  __syncthreads();

  float* sHw = sH + w * 16 * SX_STRIDE;
  const float* tfrow = tffold + (size_t)b * EE;

  // ---- GEMM1: H = lrelu(X @ W1top + tffold) ----
  for (int ct = 0; ct < 8; ++ct) {
    float init = tfrow[ct * 16 + m];   // same N for every C VGPR of this lane
    v8f acc;
#pragma unroll
    for (int g = 0; g < 8; ++g) acc[g] = init;
#pragma unroll 4
    for (int kk = 0; kk < 32; ++kk) {
      int k0 = kk * 4 + 2 * half;
      v2f a, bmat;
      a.x    = sXw[m * SX_STRIDE + k0];
      a.y    = sXw[m * SX_STRIDE + k0 + 1];
      bmat.x = sW[k0 * SW_STRIDE + ct * 16 + m];
      bmat.y = sW[(k0 + 1) * SW_STRIDE + ct * 16 + m];
      acc = __builtin_amdgcn_wmma_f32_16x16x4_f32(
          /*neg_a=*/false, a, /*neg_b=*/false, bmat,
          /*c_mod=*/(short)0, acc, /*reuse_a=*/false, /*reuse_b=*/false);
    }
#pragma unroll
    for (int g = 0; g < 8; ++g) {
      float x = acc[g];
      x = x > 0.f ? x : 0.1f * x;                       // LeakyReLU(0.1)
      sHw[(g + 8 * half) * SX_STRIDE + ct * 16 + m] = x;
    }
  }
  __syncthreads();

  // swap W2 into LDS
  for (int i = tid; i < EE * EE; i += blockDim.x) {
    int r = i >> 7, c = i & 127;
    sW[r * SW_STRIDE + c] = w2[i];
  }
  __syncthreads();

  // ---- GEMM2: Y = H @ W2 + b2 ----
  for (int ct = 0; ct < 8; ++ct) {
    float init = b2[ct * 16 + m];
    v8f acc;
#pragma unroll
    for (int g = 0; g < 8; ++g) acc[g] = init;
#pragma unroll 4
    for (int kk = 0; kk < 32; ++kk) {
      int k0 = kk * 4 + 2 * half;
      v2f a, bmat;
      a.x    = sHw[m * SX_STRIDE + k0];
      a.y    = sHw[m * SX_STRIDE + k0 + 1];
      bmat.x = sW[k0 * SW_STRIDE + ct * 16 + m];
      bmat.y = sW[(k0 + 1) * SW_STRIDE + ct * 16 + m];
      acc = __builtin_amdgcn_wmma_f32_16x16x4_f32(
          false, a, false, bmat, (short)0, acc, false, false);
    }
#pragma unroll
    for (int g = 0; g < 8; ++g)                         // stage into sXw (done with X)
      sXw[(g + 8 * half) * SX_STRIDE + ct * 16 + m] = acc[g];
  }
  // per-wave coalesced store of the 16x128 strip
  {
    float4* Yg = (float4*)(Y + r0 * EE);
    for (int i = lane; i < 16 * (EE / 4); i += 32) {
      int r = i >> 5, c4 = i & 31;
      Yg[r * (EE / 4) + c4] = ((const float4*)(sXw + r * SX_STRIDE))[c4];
    }
  }
}

// ---------------------------------------------------------------------------
// Phase-2 attention kernels (all tiny, L2-resident)
// ---------------------------------------------------------------------------
__global__ void dot_gp_kernel(const float* __restrict__ eS,
                              const float* __restrict__ uq,
                              const float* __restrict__ gp,
                              float* __restrict__ logits) {
  int w = threadIdx.x >> 5, lane = threadIdx.x & 31;
  int row = blockIdx.x * 8 + w;                  // [0, B*M)
  int b = row >> 11;
  float4 e = ((const float4*)(eS + (size_t)row * EE))[lane];
  float4 u = ((const float4*)(uq + (size_t)b * EE))[lane];
  float s = e.x * u.x + e.y * u.y + e.z * u.z + e.w * u.w;
#pragma unroll
  for (int off = 16; off > 0; off >>= 1) s += __shfl_xor(s, off, 32);
  if (lane == 0) logits[row] = gp[row] * s;
}

__global__ void softmax_kernel(const float* __restrict__ logits,
                               float* __restrict__ p,
                               float* __restrict__ out_logits) {
  __shared__ float red[256];
  int b = blockIdx.x, t = threadIdx.x;
  const float* l = logits + (size_t)b * MM;
  float mx = -3.4e38f;
  for (int i = t; i < MM; i += 256) mx = fmaxf(mx, l[i]);
  red[t] = mx; __syncthreads();
  for (int s2 = 128; s2 > 0; s2 >>= 1) {
    if (t < s2) red[t] = fmaxf(red[t], red[t + s2]);
    __syncthreads();
  }
  mx = red[0]; __syncthreads();
  float sm = 0.f;
  for (int i = t; i < MM; i += 256) sm += __expf(l[i] - mx);
  red[t] = sm; __syncthreads();
  for (int s2 = 128; s2 > 0; s2 >>= 1) {
    if (t < s2) red[t] += red[t + s2];
    __syncthreads();
  }
  float inv = 1.f / red[0];
  for (int i = t; i < MM; i += 256) {
    p[(size_t)b * MM + i] = __expf(l[i] - mx) * inv;
    if (out_logits) out_logits[(size_t)b * MM + i] = l[i];
  }
}

__global__ void uq_update_kernel(const float* __restrict__ eS,
                                 const float* __restrict__ p,
                                 const float* __restrict__ gp,
                                 float* __restrict__ uq) {
  int b = blockIdx.x, e = threadIdx.x;           // 128 threads, coalesced over e
  float acc = 0.f;
  const float* col = eS + (size_t)b * MM * EE + e;
  const float* pb  = p + (size_t)b * MM;
  const float* gb  = gp + (size_t)b * MM;
  for (int m = 0; m < MM; ++m) acc += pb[m] * gb[m] * col[(size_t)m * EE];
  uq[(size_t)b * EE + e] += acc;
}

__global__ void copy_kernel(const float* __restrict__ src, float* __restrict__ dst, int n) {
  int t = blockIdx.x * blockDim.x + threadIdx.x;
  if (t < n) dst[t] = src[t];
}

// ---------------------------------------------------------------------------
extern "C" void kernel_launch(void* const* d_in, const int* in_sizes, int n_in,
                              void* d_out, int out_size, void* d_ws, size_t ws_size,
                              hipStream_t stream) {
  const int*   story    = (const int*)d_in[0];
  const int*   kb_len   = (const int*)d_in[1];
  const int*   conv_len = (const int*)d_in[2];
  // d_in[3] = hidden   : dead (only feeds u, which never reaches the output)
  const float* dh       = (const float*)d_in[4];
  const float* tf       = (const float*)d_in[5];
  const float* qv       = (const float*)d_in[6];
  const float* gp       = (const float*)d_in[7];
  const float* C        = (const float*)d_in[8];
  const float* wA1      = (const float*)d_in[9];
  const float* bA1      = (const float*)d_in[10];
  const float* wA2      = (const float*)d_in[11];
  const float* bA2      = (const float*)d_in[12];
  const float* wC1      = (const float*)d_in[13];
  const float* bC1      = (const float*)d_in[14];
  const float* wC2      = (const float*)d_in[15];
  const float* bC2      = (const float*)d_in[16];
  // d_in[17]=wf, d_in[18]=bf : dead

  float* ws     = (float*)d_ws;
  float* base   = ws;                                   // B*M*E       (reused per k)
  float* mst    = base + (size_t)BB * MM * EE;          // 4 * B*M*E   (m_story)
  float* tfA    = mst + (size_t)4 * BB * MM * EE;       // B*E
  float* tfC    = tfA + BB * EE;                        // B*E
  float* uq     = tfC + BB * EE;                        // B*E
  float* logits = uq + BB * EE;                         // B*M
  float* probs  = logits + BB * MM;                     // B*M
  float* outp   = (float*)d_out;                        // [prob_soft | prob_logits]

  // fold tf half of concat into per-batch bias for both weight sets
  tf_fold_kernel<<<(BB * EE) / 256, 256, 0, stream>>>(tf, wA1, bA1, tfA);
  tf_fold_kernel<<<(BB * EE) / 256, 256, 0, stream>>>(tf, wC1, bC1, tfC);

  const size_t smem_bytes =
      (size_t)(EE * SW_STRIDE + 2 * MLP_WAVES * 16 * SX_STRIDE) * sizeof(float);

  // m_story = [MLP_A(base0), MLP_A(base1), MLP_A(base2), MLP_C(base3)]
  for (int k = 0; k < 4; ++k) {
    emb_addlm_kernel<<<BB * MM / 8, 256, 0, stream>>>(
        story, kb_len, conv_len, dh, C + (size_t)k * VV * EE, base);
    const float* w1  = (k < 3) ? wA1 : wC1;
    const float* tff = (k < 3) ? tfA : tfC;
    const float* w2  = (k < 3) ? wA2 : wC2;
    const float* b2  = (k < 3) ? bA2 : bC2;
    mlp_wmma_kernel<<<BB * MM / (MLP_WAVES * 16), MLP_WAVES * 32, smem_bytes, stream>>>(
        base, w1, tff, w2, b2, mst + (size_t)k * BB * MM * EE);
  }

  // phase-2 attention with uq = query_vector
  copy_kernel<<<(BB * EE + 255) / 256, 256, 0, stream>>>(qv, uq, BB * EE);
  for (int h = 0; h < 3; ++h) {
    dot_gp_kernel<<<BB * MM / 8, 256, 0, stream>>>(
        mst + (size_t)h * BB * MM * EE, uq, gp, logits);
    if (h == 2) {
      softmax_kernel<<<BB, 256, 0, stream>>>(logits, outp, outp + BB * MM);
    } else {
      softmax_kernel<<<BB, 256, 0, stream>>>(logits, probs, nullptr);
      uq_update_kernel<<<BB, EE, 0, stream>>>(
          mst + (size_t)(h + 1) * BB * MM * EE, probs, gp, uq);
    }
  }
}